// HemisphereRaysampler_44014824849466
// MI455X (gfx1250) — compile-verified
//
#include <hip/hip_runtime.h>
#include <hip/hip_bf16.h>
#include <math.h>

// ---------------------------------------------------------------------------
// Problem constants (from reference)
// ---------------------------------------------------------------------------
#define HNUM 256
#define WNUM 256
#define MC   128
#define RR   1024           // NSQ*NSQ
#define TT   512
#define BINL 0.02f
#define EPSN 1e-12f

// d_out flat layout (float elements), outputs concatenated in return order:
//   origins_out    [0, 393216)
//   directions_out [393216, 786432)
//   t              [786432, 67895296)
//   xys            [67895296, 68157440)
#define OFF_ORIG 0
#define OFF_DIR  (MC*RR*3)                  // 393216
#define OFF_T    (2*MC*RR*3)                // 786432
#define OFF_XYS  (2*MC*RR*3 + MC*RR*TT)     // 67895296

// Scratch carved out of d_out regions that are overwritten LAST by the
// broadcast kernel (so no dependence on ws_size):
//   region A (inside origins_out):
#define S_LVEC4  0            // 128*4 floats : normalized camera->laser, pad 0
#define S_DIR4   512          // 1024*4 floats: directions padded with 0
#define S_LVAL   4608         // 128 floats  : l (|laser-camera| at pixel)
#define S_EFF    4736         // 128 floats  : eff (float)
#define S_D1D4   4864         // 128 floats  : d1+d4 at pixel
//   flags: first 128 ints of directions_out region
//   dx   : first 131072 floats of xys region

typedef float v2f __attribute__((ext_vector_type(2)));
typedef float v4f __attribute__((ext_vector_type(4)));
typedef float v8f __attribute__((ext_vector_type(8)));

// ---------------------------------------------------------------------------
// K0: per-m setup, padded direction table, zero NaN flags. One 1024-thr block.
// ---------------------------------------------------------------------------
__global__ void k0_setup(const float* __restrict__ laser_pos,
                         const float* __restrict__ camera_pos,
                         const float* __restrict__ laser_origin,
                         const float* __restrict__ camera_origin,
                         const float* __restrict__ directions,
                         const int*   __restrict__ idx_perm,
                         float* __restrict__ scrA,
                         int*   __restrict__ flags)
{
    const int tid = threadIdx.x;

    // pad directions to stride-4 with zero in lane 3 (for WMMA K=4)
    if (tid < RR) {
        scrA[S_DIR4 + tid*4 + 0] = directions[tid*3 + 0];
        scrA[S_DIR4 + tid*4 + 1] = directions[tid*3 + 1];
        scrA[S_DIR4 + tid*4 + 2] = directions[tid*3 + 2];
        scrA[S_DIR4 + tid*4 + 3] = 0.0f;
    }

    if (tid < MC) {
        flags[tid] = 0;

        int idx = idx_perm[tid];
        int y = (tid == 0) ? (HNUM/2) : (idx / WNUM);
        int x = (tid == 0) ? (WNUM/2) : (idx % WNUM);
        int p = y * WNUM + x;

        float lx = laser_pos[p*3+0], ly = laser_pos[p*3+1], lz = laser_pos[p*3+2];
        float cx = camera_pos[p*3+0], cy = camera_pos[p*3+1], cz = camera_pos[p*3+2];

        float lo0 = laser_origin[0], lo1 = laser_origin[1], lo2 = laser_origin[2];
        float co0 = camera_origin[0], co1 = camera_origin[1], co2 = camera_origin[2];

        float d1 = sqrtf((lx-lo0)*(lx-lo0) + (ly-lo1)*(ly-lo1) + (lz-lo2)*(lz-lo2));
        float d4 = sqrtf((cx-co0)*(cx-co0) + (cy-co1)*(cy-co1) + (cz-co2)*(cz-co2));
        float dd = d1 + d4;
        scrA[S_D1D4 + tid] = dd;

        float vx = lx - cx, vy = ly - cy, vz = lz - cz;
        float l  = sqrtf(vx*vx + vy*vy + vz*vz);
        scrA[S_LVAL + tid] = l;

        float inv = 1.0f / fmaxf(l, EPSN);
        scrA[S_LVEC4 + tid*4 + 0] = vx * inv;
        scrA[S_LVEC4 + tid*4 + 1] = vy * inv;
        scrA[S_LVEC4 + tid*4 + 2] = vz * inv;
        scrA[S_LVEC4 + tid*4 + 3] = 0.0f;

        // idx_offset = ceil((|lp-cp| - (d1+d4)) / BIN_LEN); eff = floor-mod T
        int o = (int)ceilf((l - dd) / BINL);
        int eff = ((o % TT) + TT) % TT;
        scrA[S_EFF + tid] = (float)eff;
    }
}

// ---------------------------------------------------------------------------
// K1: dx[m][r] = dot(lvec[m], dir[r]) via V_WMMA_F32_16X16X4_F32.
// One 16(M)x16(N) tile per wave32; 8x64 = 512 tiles -> 64 blocks x 8 waves.
// A (16x4 f32): VGPR0 = K0 (lanes 0-15) / K2 (lanes 16-31); VGPR1 = K1 / K3.
// B (4x16 f32): mirrored K striping. D: vgpr j -> M=j (lo half) / M=j+8 (hi).
// ---------------------------------------------------------------------------
__global__ void k1_dx_wmma(const float* __restrict__ scrA,
                           float* __restrict__ dxout)
{
    const int lane = threadIdx.x & 31;
    const int wave = threadIdx.x >> 5;
    const int tile = blockIdx.x * 8 + wave;      // 0..511
    const int mt = tile >> 6;                    // 0..7   (M tiles)
    const int rt = tile & 63;                    // 0..63  (N tiles)
    const int mbase = mt * 16;
    const int rbase = rt * 16;
    const int l15 = lane & 15;
    const int hi  = lane >> 4;                   // 0 or 1

    // A: lvec (padded to K=4 with 0)
    const int m = mbase + l15;
    v2f a;
    a[0] = scrA[S_LVEC4 + m*4 + hi*2 + 0];
    a[1] = scrA[S_LVEC4 + m*4 + hi*2 + 1];

    // B: directions (padded to K=4 with 0)
    const int r = rbase + l15;
    v2f b;
    b[0] = scrA[S_DIR4 + r*4 + hi*2 + 0];
    b[1] = scrA[S_DIR4 + r*4 + hi*2 + 1];

    v8f c = {};
    c = __builtin_amdgcn_wmma_f32_16x16x4_f32(
            /*neg_a=*/false, a, /*neg_b=*/false, b,
            /*c_mod=*/(short)0, c, /*reuse_a=*/false, /*reuse_b=*/false);

#pragma unroll
    for (int j = 0; j < 8; ++j) {
        dxout[(mbase + hi*8 + j) * RR + rbase + l15] = c[j];
    }
}

// ---------------------------------------------------------------------------
// K2: streaming pass over t. One (m,r) per 128-thread block; each thread
// computes 4 consecutive k-bins and writes one non-temporal b128 store
// (t is write-once, 256 MB > 192 MB L2 -> NT keeps L2 clean).
// OR-reduces per-m NaN flags via shared mem + one global atomic.
// ---------------------------------------------------------------------------
__global__ void k2_t_main(const float* __restrict__ scrA,
                          const float* __restrict__ dxarr,
                          int*   __restrict__ flags,
                          float* __restrict__ tout)
{
    const int b = blockIdx.x;
    const int m = b >> 10;
    const int r = b & (RR - 1);

    const float l   = scrA[S_LVAL + m];
    const float eff = scrA[S_EFF  + m];
    const float dxv = dxarr[m * RR + r];

    // faithful to reference: dy = sqrt(1-dx^2); dy^2 is NaN if dx^2 > 1
    const float dy  = sqrtf(1.0f - dxv*dxv);
    const float dy2 = dy * dy;
    const float dx2 = dxv * dxv;
    const float l2  = l * l;
    const float dxl = dxv * l;

    __shared__ int snan;
    if (threadIdx.x == 0) snan = 0;
    __syncthreads();

    const int k0 = threadIdx.x * 4;
    v4f vv;
    bool nan = false;
#pragma unroll
    for (int c = 0; c < 4; ++c) {
        const int k = k0 + c;
        const float tau  = BINL * fmaxf((float)k, eff);
        const float tau2 = tau * tau;
        const float diff = tau2 - l2;
        const float num  = (dxl + tau) * diff;
        const float den  = 2.0f * (dx2 * diff + dy2 * tau2);
        const float tn   = num / den;
        vv[c] = tn;
        nan = nan || (tn != tn);
    }
    const size_t base = ((size_t)(m * RR + r)) * TT;
    __builtin_nontemporal_store(vv, (v4f*)(tout + base + k0));

    if (nan) atomicOr(&snan, 1);
    __syncthreads();
    if (threadIdx.x == 0 && snan) atomicOr(&flags[m], 1);
}

// ---------------------------------------------------------------------------
// K3: fixup — rows m with any NaN get t_conf[m][k] broadcast over r.
// ---------------------------------------------------------------------------
__global__ void k3_fixup(const float* __restrict__ scrA,
                         const int*   __restrict__ flags,
                         float* __restrict__ tout)
{
    const int b = blockIdx.x;
    const int m = b >> 10;
    if (!flags[m]) return;
    const int r = b & (RR - 1);
    const float dd = scrA[S_D1D4 + m];

    const int k0 = threadIdx.x * 4;
    v4f vv;
#pragma unroll
    for (int c = 0; c < 4; ++c) {
        float d2d3 = BINL * (float)(k0 + c) - dd;
        if (d2d3 <= 0.0f) d2d3 = 1e-6f;
        vv[c] = 0.5f * d2d3;
    }
    const size_t base = ((size_t)(m * RR + r)) * TT;
    __builtin_nontemporal_store(vv, (v4f*)(tout + base + k0));
}

// ---------------------------------------------------------------------------
// K4: broadcast outputs (runs LAST: overwrites all scratch regions).
// Reads only original inputs — no scratch reads.
// ---------------------------------------------------------------------------
__global__ void k4_broadcast(const float* __restrict__ camera_pos,
                             const float* __restrict__ directions,
                             const int*   __restrict__ idx_perm,
                             float* __restrict__ out)
{
    const int t = blockIdx.x * blockDim.x + threadIdx.x;   // 0..131071
    const int m = t >> 10;
    const int r = t & (RR - 1);

    int idx = idx_perm[m];
    int y = (m == 0) ? (HNUM/2) : (idx / WNUM);
    int x = (m == 0) ? (WNUM/2) : (idx % WNUM);
    int p = y * WNUM + x;

    const int mr = m * RR + r;
#pragma unroll
    for (int c = 0; c < 3; ++c) {
        out[OFF_ORIG + mr*3 + c] = camera_pos[p*3 + c];
        out[OFF_DIR  + mr*3 + c] = directions[r*3 + c];
    }
    v2f yx;
    yx[0] = (float)y;
    yx[1] = (float)x;
    *(v2f*)(out + OFF_XYS + mr*2) = yx;
}

// ---------------------------------------------------------------------------
// launch
// ---------------------------------------------------------------------------
extern "C" void kernel_launch(void* const* d_in, const int* in_sizes, int n_in,
                              void* d_out, int out_size, void* d_ws, size_t ws_size,
                              hipStream_t stream)
{
    (void)in_sizes; (void)n_in; (void)out_size; (void)d_ws; (void)ws_size;

    const float* laser_pos     = (const float*)d_in[0];
    const float* camera_pos    = (const float*)d_in[1];
    const float* laser_origin  = (const float*)d_in[2];
    const float* camera_origin = (const float*)d_in[3];
    const float* directions    = (const float*)d_in[4];
    const int*   idx_perm      = (const int*)  d_in[5];

    float* out   = (float*)d_out;
    float* scrA  = out + OFF_ORIG;                // scratch inside origins_out
    int*   flags = (int*)(out + OFF_DIR);         // scratch inside directions_out
    float* dxarr = out + OFF_XYS;                 // scratch inside xys
    float* tout  = out + OFF_T;

    k0_setup<<<1, 1024, 0, stream>>>(laser_pos, camera_pos, laser_origin,
                                     camera_origin, directions, idx_perm,
                                     scrA, flags);

    k1_dx_wmma<<<64, 256, 0, stream>>>(scrA, dxarr);

    k2_t_main<<<MC * RR, 128, 0, stream>>>(scrA, dxarr, flags, tout);

    k3_fixup<<<MC * RR, 128, 0, stream>>>(scrA, flags, tout);

    k4_broadcast<<<(MC * RR) / 256, 256, 0, stream>>>(camera_pos, directions,
                                                      idx_perm, out);
}